// GCNnet_58377195487652
// MI455X (gfx1250) — compile-verified
//
#include <hip/hip_runtime.h>
#include <hip/hip_bf16.h>

typedef __attribute__((ext_vector_type(2))) float v2f;
typedef __attribute__((ext_vector_type(8))) float v8f;

// ---------------------------------------------------------------------------
// Elementwise / normalization kernels
// ---------------------------------------------------------------------------

__global__ void k_fill(float* __restrict__ p, float v, int n) {
    int i = blockIdx.x * blockDim.x + threadIdx.x;
    if (i < n) p[i] = v;
}

__global__ void k_deg_accum(const int* __restrict__ col, const float* __restrict__ w,
                            float* __restrict__ deg, int E) {
    int e = blockIdx.x * blockDim.x + threadIdx.x;
    if (e < E) unsafeAtomicAdd(&deg[col[e]], w[e]);
}

// deg -> dinv (in place), and self-loop norm = dinv^2
__global__ void k_dinv(float* __restrict__ deg, float* __restrict__ nself, int N) {
    int i = blockIdx.x * blockDim.x + threadIdx.x;
    if (i >= N) return;
    float d = deg[i];
    float r = (d > 0.0f) ? rsqrtf(d) : 0.0f;
    deg[i] = r;
    nself[i] = r * r;
}

__global__ void k_edge_norm(const int* __restrict__ row, const int* __restrict__ col,
                            const float* __restrict__ w, const float* __restrict__ dinv,
                            float* __restrict__ nedge, int E) {
    int e = blockIdx.x * blockDim.x + threadIdx.x;
    if (e >= E) return;
    nedge[e] = dinv[row[e]] * w[e] * dinv[col[e]];
}

// ---------------------------------------------------------------------------
// FP32 WMMA GEMM: C[M x N] = A[M x K] * B[K x N], row-major.
// Requirements (all met by this net): K % 8 == 0. M,N arbitrary via clamping:
//  - A row index clamped to M-1: OOB A rows only affect OOB C rows (dropped at store)
//  - B column clamped to N-1, loaded value selected to 0 (no EXEC branching)
// One wave per 16x16 C tile; two V_WMMA_F32_16X16X4_F32 per iteration.
// ---------------------------------------------------------------------------

__global__ void k_gemm_wmma_f32(const float* __restrict__ A, const float* __restrict__ B,
                                float* __restrict__ C, int M, int N, int K) {
    const int lane = threadIdx.x & 31;
    const int wave = threadIdx.x >> 5;
    const int ntn  = (N + 15) >> 4;
    const int ntm  = (M + 15) >> 4;
    const int tile = blockIdx.x * (blockDim.x >> 5) + wave;
    if (tile >= ntm * ntn) return;           // wave-uniform: EXEC all-1s at WMMA
    const int tm = (tile / ntn) << 4;
    const int tn = (tile % ntn) << 4;

    const int half = lane >> 4;              // 0: lanes 0-15, 1: lanes 16-31
    const int l    = lane & 15;

    // A fragment base: row tm+l (clamped), this half covers K offsets {0,1}/{2,3}
    const int arow = min(tm + l, M - 1);
    const float* ap = A + (size_t)arow * K + half * 2;

    // B fragment base: column tn+l (clamped+masked), rows k+{0,1}/{2,3}
    const int  bcol = tn + l;
    const bool cok  = bcol < N;
    const int  bc   = cok ? bcol : (N - 1);
    const float* bp = B + (size_t)(half * 2) * N + bc;

    v8f c = {};
    for (int k = 0; k < K; k += 8) {
        const v2f a0 = *(const v2f*)(ap + k);        // 8B vector load
        const v2f a1 = *(const v2f*)(ap + k + 4);
        const float t0 = bp[(size_t)k * N];
        const float t1 = bp[(size_t)k * N + N];
        const float t2 = bp[(size_t)(k + 4) * N];
        const float t3 = bp[(size_t)(k + 4) * N + N];
        v2f b0, b1;
        b0.x = cok ? t0 : 0.0f;  b0.y = cok ? t1 : 0.0f;
        b1.x = cok ? t2 : 0.0f;  b1.y = cok ? t3 : 0.0f;
        c = __builtin_amdgcn_wmma_f32_16x16x4_f32(false, a0, false, b0,
                                                  (short)0, c, false, false);
        c = __builtin_amdgcn_wmma_f32_16x16x4_f32(false, a1, false, b1,
                                                  (short)0, c, false, false);
    }

    // C layout: VGPR v -> row tm+v (lanes 0-15) / tm+v+8 (lanes 16-31), col tn+l
    #pragma unroll
    for (int v = 0; v < 8; ++v) {
        const int r = tm + v + half * 8;
        if (r < M && cok) C[(size_t)r * N + bcol] = c[v];
    }
}

// ---------------------------------------------------------------------------
// GCN message passing (F % 4 == 0 always: 40/80/128/512 -> float4 traffic)
// ---------------------------------------------------------------------------

// acc[n][f] = hw[n][f] * nself[n]  (folds self-loop; fully overwrites poison)
__global__ void k_init_acc4(const float* __restrict__ hw, const float* __restrict__ nself,
                            float* __restrict__ acc, int total4, int Fq) {
    int i = blockIdx.x * blockDim.x + threadIdx.x;
    if (i >= total4) return;
    const int n = i / Fq;
    const float s = nself[n];
    const float4 v = ((const float4*)hw)[i];
    float4 o; o.x = v.x * s; o.y = v.y * s; o.z = v.z * s; o.w = v.w * s;
    ((float4*)acc)[i] = o;
}

// acc[col[e]][f..f+3] += hw[row[e]][f..f+3] * nedge[e]
__global__ void k_scatter4(const float* __restrict__ hw, const int* __restrict__ row,
                           const int* __restrict__ col, const float* __restrict__ nedge,
                           float* __restrict__ acc, int total4, int Fq) {
    int i = blockIdx.x * blockDim.x + threadIdx.x;
    if (i >= total4) return;
    const int e  = i / Fq;
    const int f4 = (i - e * Fq) * 4;
    const int F  = Fq * 4;
    const int   r = __builtin_nontemporal_load(row + e);
    const int   cdst = __builtin_nontemporal_load(col + e);
    const float nrm  = __builtin_nontemporal_load(nedge + e);
    const float4 v = *(const float4*)(hw + (size_t)r * F + f4);
    float* dst = acc + (size_t)cdst * F + f4;
    unsafeAtomicAdd(dst + 0, v.x * nrm);
    unsafeAtomicAdd(dst + 1, v.y * nrm);
    unsafeAtomicAdd(dst + 2, v.z * nrm);
    unsafeAtomicAdd(dst + 3, v.w * nrm);
}

__global__ void k_bias_relu4(float* __restrict__ h, const float* __restrict__ b,
                             int total4, int Fq) {
    int i = blockIdx.x * blockDim.x + threadIdx.x;
    if (i >= total4) return;
    const int f4 = (i % Fq) * 4;
    float4 v = ((const float4*)h)[i];
    v.x = fmaxf(v.x + b[f4 + 0], 0.0f);
    v.y = fmaxf(v.y + b[f4 + 1], 0.0f);
    v.z = fmaxf(v.z + b[f4 + 2], 0.0f);
    v.w = fmaxf(v.w + b[f4 + 3], 0.0f);
    ((float4*)h)[i] = v;
}

// segment max over sorted batch: values >= 0 after ReLU -> int-bit atomicMax exact
__global__ void k_pool_max4(const float* __restrict__ h, const int* __restrict__ batch,
                            float* __restrict__ g, int total4, int Fq) {
    int i = blockIdx.x * blockDim.x + threadIdx.x;
    if (i >= total4) return;
    const int n  = i / Fq;
    const int f4 = (i - n * Fq) * 4;
    const int F  = Fq * 4;
    const float4 v = ((const float4*)h)[i];
    int* dst = (int*)(g + (size_t)batch[n] * F + f4);
    atomicMax(dst + 0, __float_as_int(v.x));
    atomicMax(dst + 1, __float_as_int(v.y));
    atomicMax(dst + 2, __float_as_int(v.z));
    atomicMax(dst + 3, __float_as_int(v.w));
}

// logits = g @ Wfc2 + bfc2 ; softmax over 2 classes
__global__ void k_fc2_softmax(const float* __restrict__ g, const float* __restrict__ W,
                              const float* __restrict__ b, float* __restrict__ out,
                              int G, int K) {
    int r = blockIdx.x * blockDim.x + threadIdx.x;
    if (r >= G) return;
    float l0 = b[0], l1 = b[1];
    const float* gr = g + (size_t)r * K;
    for (int k = 0; k < K; ++k) {
        const float v = gr[k];
        l0 = fmaf(v, W[2 * k],     l0);
        l1 = fmaf(v, W[2 * k + 1], l1);
    }
    const float m  = fmaxf(l0, l1);
    const float e0 = __expf(l0 - m);
    const float e1 = __expf(l1 - m);
    const float s  = e0 + e1;
    out[2 * r]     = e0 / s;
    out[2 * r + 1] = e1 / s;
}

// ---------------------------------------------------------------------------
// Host-side orchestration
// ---------------------------------------------------------------------------

static inline int gridFor(long long total, int block) {
    return (int)((total + block - 1) / block);
}

extern "C" void kernel_launch(void* const* d_in, const int* in_sizes, int n_in,
                              void* d_out, int out_size, void* d_ws, size_t ws_size,
                              hipStream_t stream) {
    const float* x     = (const float*)d_in[0];
    const int*   eidx  = (const int*)  d_in[1];
    const float* ew    = (const float*)d_in[2];
    const int*   batch = (const int*)  d_in[3];
    const float* W1    = (const float*)d_in[4];
    const float* b1    = (const float*)d_in[5];
    const float* W2    = (const float*)d_in[6];
    const float* b2    = (const float*)d_in[7];
    const float* W3    = (const float*)d_in[8];
    const float* b3    = (const float*)d_in[9];
    const float* Wfc1  = (const float*)d_in[10];
    const float* bfc1  = (const float*)d_in[11];
    const float* Wfc2  = (const float*)d_in[12];
    const float* bfc2  = (const float*)d_in[13];

    const int E    = in_sizes[2];        // 1,600,000 edges
    const int N    = in_sizes[3];        // 100,000 nodes
    const int F0   = in_sizes[5];        // 40
    const int F1   = in_sizes[7];        // 80
    const int F2   = in_sizes[9];        // 128
    const int H    = in_sizes[11];       // 512
    const int G    = out_size / 2;       // 512 graphs
    const int FMAX = F2;

    const int* erow = eidx;              // edge_index[0]
    const int* ecol = eidx + E;          // edge_index[1]

    // workspace carve-up (floats; all offsets multiples of 4 -> float4 aligned)
    float* w = (float*)d_ws;
    float* dinv  = w;  w += N;                       // deg -> dinv (in place)
    float* nself = w;  w += N;
    float* nedge = w;  w += E;
    float* hw    = w;  w += (size_t)N * FMAX;        // GEMM output
    float* acc   = w;  w += (size_t)N * FMAX;        // scatter accumulator / next h
    float* g     = w;  w += (size_t)G * F2;          // pooled graph features
    float* fc1   = w;  w += (size_t)G * H;           // FC1 activations
    (void)ws_size; (void)n_in;

    const int BLK = 256;

    // ---- gcn_norm ----
    k_fill<<<gridFor(N, BLK), BLK, 0, stream>>>(dinv, 1.0f, N);   // self-loop weight
    k_deg_accum<<<gridFor(E, BLK), BLK, 0, stream>>>(ecol, ew, dinv, E);
    k_dinv<<<gridFor(N, BLK), BLK, 0, stream>>>(dinv, nself, N);
    k_edge_norm<<<gridFor(E, BLK), BLK, 0, stream>>>(erow, ecol, ew, dinv, nedge, E);

    // ---- 3 GCN layers ----
    const float* layerA[3] = { x,  acc, acc };
    const float* layerW[3] = { W1, W2,  W3  };
    const float* layerB[3] = { b1, b2,  b3  };
    const int    layerK[3] = { F0, F0,  F1  };
    const int    layerN[3] = { F0, F1,  F2  };

    for (int li = 0; li < 3; ++li) {
        const int Kd = layerK[li], Nd = layerN[li];
        const int tiles = ((N + 15) / 16) * ((Nd + 15) / 16);
        k_gemm_wmma_f32<<<gridFor(tiles, 8), BLK, 0, stream>>>(layerA[li], layerW[li],
                                                               hw, N, Nd, Kd);
        const int Fq     = Nd / 4;
        const int tot4   = N * Fq;
        const long long e4 = (long long)E * Fq;
        k_init_acc4<<<gridFor(tot4, BLK), BLK, 0, stream>>>(hw, nself, acc, tot4, Fq);
        k_scatter4<<<gridFor(e4, BLK), BLK, 0, stream>>>(hw, erow, ecol, nedge, acc,
                                                         (int)e4, Fq);
        k_bias_relu4<<<gridFor(tot4, BLK), BLK, 0, stream>>>(acc, layerB[li], tot4, Fq);
    }

    // ---- global max pool (post-ReLU values >= 0 -> int-bit max is exact) ----
    k_fill<<<gridFor((long long)G * F2, BLK), BLK, 0, stream>>>(g, 0.0f, G * F2);
    k_pool_max4<<<gridFor((long long)N * (F2 / 4), BLK), BLK, 0, stream>>>(
        acc, batch, g, N * (F2 / 4), F2 / 4);

    // ---- FC1 (WMMA) + bias/ReLU ----
    {
        const int tiles = ((G + 15) / 16) * ((H + 15) / 16);
        k_gemm_wmma_f32<<<gridFor(tiles, 8), BLK, 0, stream>>>(g, Wfc1, fc1, G, H, F2);
        k_bias_relu4<<<gridFor((long long)G * (H / 4), BLK), BLK, 0, stream>>>(
            fc1, bfc1, G * (H / 4), H / 4);
    }

    // ---- FC2 + softmax ----
    k_fc2_softmax<<<gridFor(G, BLK), BLK, 0, stream>>>(fc1, Wfc2, bfc2,
                                                       (float*)d_out, G, H);
}